// TokenCounter_78099685310872
// MI455X (gfx1250) — compile-verified
//
#include <hip/hip_runtime.h>
#include <hip/hip_bf16.h>

// TokenCounter for MI455X (gfx1250, wave32).
//
// Reference: T=1M tokens x K=8 expert indices, E=64 experts.
// counts[e] = #tokens whose index row contains e (duplicates in a row count once).
//
// Bandwidth-bound: 33.5 MB of int32 indices @ 23.3 TB/s => ~1.4 us floor.
// Strategy: 128-bit non-temporal streaming loads, per-token 64-bit membership
// mask (exact set semantics), LDS-resident 64-bin histogram via ds_add_u32
// (no-return LDS atomics), per-block partials in d_ws, tiny deterministic
// reduction kernel -> 64 floats in d_out. No global atomics, no reliance on
// prior buffer contents.

#define NBINS 64
#define BS    256   // 8 wave32 waves per block
#define MAXNB 1024

// Native clang vector type: accepted by __builtin_nontemporal_load,
// lowers to a single global_load_b128 (NT hint).
typedef int v4i __attribute__((ext_vector_type(4)));

__global__ __launch_bounds__(BS)
void TokenCounter_hist_kernel(const int* __restrict__ indices,
                              int* __restrict__ partials,
                              int T) {
    __shared__ int bins[NBINS];
    const int tid = threadIdx.x;
    if (tid < NBINS) bins[tid] = 0;
    __syncthreads();

    // 8 int32 indices per token = two 16B chunks; stream with NT hint.
    const v4i* __restrict__ p = (const v4i*)indices;
    const int stride = gridDim.x * BS;

    for (int t = blockIdx.x * BS + tid; t < T; t += stride) {
        const v4i a = __builtin_nontemporal_load(&p[2 * t + 0]);
        const v4i b = __builtin_nontemporal_load(&p[2 * t + 1]);

        // 64-bit membership mask: dedup within the token's row (set semantics).
        unsigned long long m =
            (1ull << (a.x & 63)) | (1ull << (a.y & 63)) |
            (1ull << (a.z & 63)) | (1ull << (a.w & 63)) |
            (1ull << (b.x & 63)) | (1ull << (b.y & 63)) |
            (1ull << (b.z & 63)) | (1ull << (b.w & 63));

        // Walk set bits (<= 8 total) with cheap 32-bit ops; ds_add_u32 per bit.
        unsigned lo = (unsigned)m;
        unsigned hi = (unsigned)(m >> 32);
        while (lo) {
            const int e = __builtin_ctz(lo);
            lo &= lo - 1;
            atomicAdd(&bins[e], 1);
        }
        while (hi) {
            const int e = __builtin_ctz(hi);
            hi &= hi - 1;
            atomicAdd(&bins[32 + e], 1);
        }
    }

    __syncthreads();
    if (tid < NBINS) partials[blockIdx.x * NBINS + tid] = bins[tid];
}

__global__ __launch_bounds__(BS)
void TokenCounter_reduce_kernel(const int* __restrict__ partials,
                                float* __restrict__ out,
                                int nb) {
    __shared__ int acc[BS];
    const int e = threadIdx.x & (NBINS - 1);
    const int q = threadIdx.x >> 6;   // 0..3
    int s = 0;
    // Threads with the same q read a contiguous 256B row of partials: coalesced.
    for (int b = q; b < nb; b += 4) {
        s += partials[b * NBINS + e];
    }
    acc[threadIdx.x] = s;
    __syncthreads();
    if (threadIdx.x < NBINS) {
        const int tot = acc[e] + acc[e + 64] + acc[e + 128] + acc[e + 192];
        out[e] = (float)tot;  // counts <= 2^20: exact in f32
    }
}

extern "C" void kernel_launch(void* const* d_in, const int* in_sizes, int n_in,
                              void* d_out, int out_size, void* d_ws, size_t ws_size,
                              hipStream_t stream) {
    (void)n_in; (void)out_size;
    const int* indices = (const int*)d_in[0];     // [T, 8] int32
    const int  TK      = in_sizes[0];
    const int  T       = TK / 8;

    int* partials = (int*)d_ws;
    // One 64-int partial histogram per block; clamp block count to workspace.
    long long nb_ll = (long long)(ws_size / (NBINS * sizeof(int)));
    int nb = (nb_ll > MAXNB) ? MAXNB : (int)nb_ll;
    if (nb < 1) nb = 1;
    // No more blocks than needed (>=1 token per thread).
    const int max_useful = (T + BS - 1) / BS;
    if (nb > max_useful) nb = max_useful;
    if (nb < 1) nb = 1;

    TokenCounter_hist_kernel<<<nb, BS, 0, stream>>>(indices, partials, T);
    TokenCounter_reduce_kernel<<<1, BS, 0, stream>>>(partials, (float*)d_out, nb);
}